// TangentProjector_87239375716828
// MI455X (gfx1250) — compile-verified
//
#include <hip/hip_runtime.h>
#include <hip/hip_bf16.h>
#include <math.h>
#include <stdint.h>

// Problem constants (from reference)
#define BATCH 32
#define NODES 512
#define NEDGE 1024      // E: matrix dimension for Cholesky
#define EPS   1e-6f

typedef __attribute__((ext_vector_type(2))) float v2f;
typedef __attribute__((ext_vector_type(8))) float v8f;
typedef __attribute__((ext_vector_type(4))) unsigned int u32x4;
typedef __attribute__((ext_vector_type(8))) int i32x8;
typedef __attribute__((ext_vector_type(4))) int i32x4;

#if __has_builtin(__builtin_amdgcn_tensor_load_to_lds)
#define TP_HAVE_TDM 1
#endif

// ---------------------------------------------------------------------------
// Kernel 1: per-edge geometry: diff_x[b][e][3], rhs[b][e] = 2*dot(dx, dv)
// ---------------------------------------------------------------------------
__global__ __launch_bounds__(256)
void tp_edge_kernel(const float* __restrict__ x, const float* __restrict__ v,
                    const int* __restrict__ ei,
                    float* __restrict__ diff, float* __restrict__ rhs) {
    int idx = blockIdx.x * blockDim.x + threadIdx.x;
    if (idx >= BATCH * NEDGE) return;
    int b = idx / NEDGE, e = idx % NEDGE;
    int i = ei[e], j = ei[NEDGE + e];
    const float* xb = x + (size_t)b * NODES * 3;
    const float* vb = v + (size_t)b * NODES * 3;
    float dx0 = xb[i*3+0] - xb[j*3+0];
    float dx1 = xb[i*3+1] - xb[j*3+1];
    float dx2 = xb[i*3+2] - xb[j*3+2];
    float dv0 = vb[i*3+0] - vb[j*3+0];
    float dv1 = vb[i*3+1] - vb[j*3+1];
    float dv2 = vb[i*3+2] - vb[j*3+2];
    float* d = diff + (size_t)idx * 3;
    d[0] = dx0; d[1] = dx1; d[2] = dx2;
    rhs[idx] = 2.0f * (dx0*dv0 + dx1*dv1 + dx2*dv2);
}

// ---------------------------------------------------------------------------
// Kernel 2: build A[b] = 4 * (diff_x diff_x^T) * sign + eps*I via f32 WMMA.
// One 16x16 output tile per wave (32 threads). K=3 padded to 4 with zeros.
// A-frag (16x4 f32, 2 VGPRs): lane l holds row M=l&15, K = (l<16)?{0,1}:{2,3}
// B-frag (4x16 f32, 2 VGPRs): same lane mapping with f-tile rows (B = Dx_f^T)
// C/D (16x16 f32, 8 VGPRs): vgpr r, lane l -> M = r + (l<16?0:8), N = l&15
// ---------------------------------------------------------------------------
__global__ __launch_bounds__(32)
void tp_buildA_kernel(const float* __restrict__ diff, const int* __restrict__ ei,
                      float* __restrict__ Ag) {
    const int tf = blockIdx.x;        // col tile
    const int te = blockIdx.y;        // row tile
    const int b  = blockIdx.z;
    const int lane = threadIdx.x;
    const int m  = lane & 15;
    const int kk = (lane < 16) ? 0 : 2;

    const float* db = diff + (size_t)b * NEDGE * 3;

    v2f a, bf;
    a.x  = db[(te*16 + m)*3 + kk];
    a.y  = (kk + 1 < 3) ? db[(te*16 + m)*3 + kk + 1] : 0.0f;
    bf.x = db[(tf*16 + m)*3 + kk];
    bf.y = (kk + 1 < 3) ? db[(tf*16 + m)*3 + kk + 1] : 0.0f;

    v8f acc = {};
    acc = __builtin_amdgcn_wmma_f32_16x16x4_f32(false, a, false, bf,
                                                (short)0, acc, false, false);

    // Apply sign mask and eps*I, store.
    int fcol = tf*16 + (lane & 15);
    int fi = ei[fcol], fj = ei[NEDGE + fcol];
    float* Ab = Ag + (size_t)b * NEDGE * NEDGE;
    #pragma unroll
    for (int r = 0; r < 8; ++r) {
        int erow = te*16 + r + ((lane < 16) ? 0 : 8);
        int eiI = ei[erow], eiJ = ei[NEDGE + erow];
        float sgn = (float)(eiI == fi) - (float)(eiI == fj)
                  - (float)(eiJ == fi) + (float)(eiJ == fj);
        float val = 4.0f * acc[r] * sgn + ((erow == fcol) ? EPS : 0.0f);
        Ab[(size_t)erow * NEDGE + fcol] = val;
    }
}

// ---------------------------------------------------------------------------
// Kernel 3: blocked Cholesky (lower), one batch per workgroup (8 waves).
// Panel width NB=16. Panel column staged in LDS; trailing SYRK tiles via
// V_WMMA_F32_16X16X4_F32 accumulating S = P_i * P_j^T, then A -= S (VALU).
// Diagonal 16x16 tile is staged into LDS by the Tensor Data Mover
// (tensor_load_to_lds, tracked on TENSORcnt) when available.
// LDS: panel 1008x16 f32 (63KB) + diag 16x16 (1KB) = 64KB.
// ---------------------------------------------------------------------------
__global__ __launch_bounds__(256)
void tp_cholesky_kernel(float* __restrict__ Ag) {
    __shared__ float s_panel[1008 * 16];
    __shared__ float s_diag[16 * 16];

    const int b   = blockIdx.x;
    const int tid = threadIdx.x;
    const int lane = tid & 31;
    const int wave = tid >> 5;
    float* A = Ag + (size_t)b * NEDGE * NEDGE;

    const int NT = NEDGE / 16;  // 64 panel steps
    for (int k = 0; k < NT; ++k) {
        const int k16 = k * 16;

        // --- stage diagonal block into LDS ---
#if defined(TP_HAVE_TDM)
        if (wave == 0) {
            // Tensor DMA descriptor (D#): 2D tile 16x16 f32 out of a
            // 1024x1024 f32 tensor, row stride 1024 elements.
            unsigned long long gaddr =
                (unsigned long long)(uintptr_t)(A + (size_t)k16 * NEDGE + k16);
            unsigned lds_addr = (unsigned)(uintptr_t)(void*)s_diag;
            u32x4 g0;
            g0[0] = 1u;                                   // count=1, user desc
            g0[1] = lds_addr;                             // lds_addr[31:0]
            g0[2] = (unsigned)gaddr;                      // global_addr[31:0]
            g0[3] = ((unsigned)(gaddr >> 32) & 0x01FFFFFFu) | 0x80000000u; // addr hi | type=2
            i32x8 g1 = {};
            g1[0] = 0x20000;          // workgroup_mask=0, data_size=2 (4B)
            g1[1] = 0x04000000;       // tensor_dim0 = 1024 (low16 at [63:48])
            g1[2] = 0x04000000;       // tensor_dim0 hi=0; tensor_dim1=1024 (low16)
            g1[3] = 16 << 16;         // tensor_dim1 hi=0; tile_dim0 = 16
            g1[4] = 16;               // tile_dim1 = 16; tile_dim2 = 0
            g1[5] = 1024;             // tensor_dim0_stride = 1024
            g1[6] = 0x04000000;       // tensor_dim1_stride = 1024 (unused for 2D)
            i32x4 z4 = {};
#if __has_include(<hip/amd_detail/amd_gfx1250_TDM.h>)
            i32x8 z8 = {};
            __builtin_amdgcn_tensor_load_to_lds(g0, g1, z4, z4, z8, 0);
#else
            __builtin_amdgcn_tensor_load_to_lds(g0, g1, z4, z4, 0);
#endif
            __builtin_amdgcn_s_wait_tensorcnt(0);
        }
#else
        {
            int r = tid >> 4, c = tid & 15;
            s_diag[r * 16 + c] = A[(size_t)(k16 + r) * NEDGE + k16 + c];
        }
#endif
        __syncthreads();

        // --- 16x16 Cholesky of diagonal block (serial, thread 0) ---
        if (tid == 0) {
            for (int c = 0; c < 16; ++c) {
                float d = sqrtf(s_diag[c * 16 + c]);
                s_diag[c * 16 + c] = d;
                float inv = 1.0f / d;
                for (int r = c + 1; r < 16; ++r) s_diag[r * 16 + c] *= inv;
                for (int r = c + 1; r < 16; ++r) {
                    float lrc = s_diag[r * 16 + c];
                    for (int q = c + 1; q <= r; ++q)
                        s_diag[r * 16 + q] -= lrc * s_diag[q * 16 + c];
                }
            }
        }
        __syncthreads();

        // --- write L_kk (lower) back to A ---
        {
            int r = tid >> 4, c = tid & 15;
            if (c <= r) A[(size_t)(k16 + r) * NEDGE + k16 + c] = s_diag[r * 16 + c];
        }

        // --- panel triangular solve: rows below diag, X * L_kk^T = A_panel ---
        for (int row = k16 + 16 + tid; row < NEDGE; row += 256) {
            float xr[16];
            #pragma unroll
            for (int c = 0; c < 16; ++c) xr[c] = A[(size_t)row * NEDGE + k16 + c];
            #pragma unroll
            for (int c = 0; c < 16; ++c) {
                float s = xr[c];
                for (int m2 = 0; m2 < c; ++m2) s -= xr[m2] * s_diag[c * 16 + m2];
                xr[c] = s / s_diag[c * 16 + c];
            }
            int lrow = row - (k16 + 16);
            #pragma unroll
            for (int c = 0; c < 16; ++c) {
                A[(size_t)row * NEDGE + k16 + c] = xr[c];
                s_panel[lrow * 16 + c] = xr[c];
            }
        }
        __syncthreads();

        // --- trailing SYRK update via WMMA: tiles (ti >= tj) below/right ---
        const int mrem = NT - (k + 1);               // remaining tile rows
        const int totalTiles = mrem * (mrem + 1) / 2;
        for (int t = wave; t < totalTiles; t += 8) {
            int ti = (int)((sqrtf(8.0f * (float)t + 1.0f) - 1.0f) * 0.5f);
            while ((ti + 1) * (ti + 2) / 2 <= t) ++ti;
            while (ti * (ti + 1) / 2 > t) --ti;
            int tj = t - ti * (ti + 1) / 2;

            const int rowBase = (k + 1 + ti) * 16;
            const int colBase = (k + 1 + tj) * 16;
            const float* pI = s_panel + ti * 16 * 16;
            const float* pJ = s_panel + tj * 16 * 16;
            __builtin_prefetch(&A[(size_t)rowBase * NEDGE + colBase], 1, 1);

            const int mloc = lane & 15;
            const int kof  = (lane < 16) ? 0 : 2;
            v8f acc = {};
            #pragma unroll
            for (int c = 0; c < 4; ++c) {
                v2f a, bb;
                a.x  = pI[mloc * 16 + c * 4 + kof];
                a.y  = pI[mloc * 16 + c * 4 + kof + 1];
                bb.x = pJ[mloc * 16 + c * 4 + kof];
                bb.y = pJ[mloc * 16 + c * 4 + kof + 1];
                acc = __builtin_amdgcn_wmma_f32_16x16x4_f32(false, a, false, bb,
                                                            (short)0, acc, false, false);
            }
            // A_tile -= acc  (f32 WMMA has no A/B negate; CNeg only)
            #pragma unroll
            for (int r = 0; r < 8; ++r) {
                int gm = rowBase + r + ((lane < 16) ? 0 : 8);
                int gn = colBase + (lane & 15);
                A[(size_t)gm * NEDGE + gn] -= acc[r];
            }
        }
        __syncthreads();
    }
}

// ---------------------------------------------------------------------------
// Kernel 4: per-batch forward (L y = rhs) then backward (L^T lam = y) solve,
// column-oriented, in-place in rhs. One workgroup per batch.
// ---------------------------------------------------------------------------
__global__ __launch_bounds__(256)
void tp_solve_kernel(const float* __restrict__ Ag, float* __restrict__ rhs) {
    const int b = blockIdx.x;
    const int tid = threadIdx.x;
    const float* A = Ag + (size_t)b * NEDGE * NEDGE;
    float* y = rhs + (size_t)b * NEDGE;

    // forward substitution
    for (int j = 0; j < NEDGE; ++j) {
        if (tid == 0) y[j] /= A[(size_t)j * NEDGE + j];
        __syncthreads();
        float yj = y[j];
        for (int i = j + 1 + tid; i < NEDGE; i += 256)
            y[i] -= A[(size_t)i * NEDGE + j] * yj;
        __syncthreads();
    }
    // backward substitution with L^T
    for (int j = NEDGE - 1; j >= 0; --j) {
        if (tid == 0) y[j] /= A[(size_t)j * NEDGE + j];
        __syncthreads();
        float yj = y[j];
        for (int i = tid; i < j; i += 256)
            y[i] -= A[(size_t)j * NEDGE + i] * yj;
        __syncthreads();
    }
}

// ---------------------------------------------------------------------------
// Kernel 5a: out = v ;  5b: scatter  out[:,i] -= 2*lam*dx ; out[:,j] += ...
// ---------------------------------------------------------------------------
__global__ __launch_bounds__(256)
void tp_initout_kernel(const float* __restrict__ v, float* __restrict__ out) {
    int idx = blockIdx.x * blockDim.x + threadIdx.x;
    if (idx < BATCH * NODES * 3) out[idx] = v[idx];
}

__global__ __launch_bounds__(256)
void tp_scatter_kernel(const float* __restrict__ lam, const float* __restrict__ diff,
                       const int* __restrict__ ei, float* __restrict__ out) {
    int idx = blockIdx.x * blockDim.x + threadIdx.x;
    if (idx >= BATCH * NEDGE) return;
    int b = idx / NEDGE, e = idx % NEDGE;
    int i = ei[e], j = ei[NEDGE + e];
    float l2 = 2.0f * lam[idx];
    const float* d = diff + (size_t)idx * 3;
    float* ob = out + (size_t)b * NODES * 3;
    #pragma unroll
    for (int dd = 0; dd < 3; ++dd) {
        float ce = l2 * d[dd];
        atomicAdd(&ob[i * 3 + dd], -ce);   // out = v - corr ; corr[i] += ce
        atomicAdd(&ob[j * 3 + dd],  ce);   // corr[j] -= ce
    }
}

// ---------------------------------------------------------------------------
extern "C" void kernel_launch(void* const* d_in, const int* in_sizes, int n_in,
                              void* d_out, int out_size, void* d_ws, size_t ws_size,
                              hipStream_t stream) {
    const float* x  = (const float*)d_in[0];
    const float* v  = (const float*)d_in[1];
    const int*   ei = (const int*)d_in[2];
    float* out = (float*)d_out;

    // Workspace layout: A (B*E*E f32 = 128 MB, L2-resident), diff (B*E*3), rhs (B*E)
    float* A    = (float*)d_ws;
    float* diff = A + (size_t)BATCH * NEDGE * NEDGE;
    float* rhs  = diff + (size_t)BATCH * NEDGE * 3;

    // 1) edge geometry
    tp_edge_kernel<<<(BATCH * NEDGE + 255) / 256, 256, 0, stream>>>(x, v, ei, diff, rhs);

    // 2) build A via WMMA f32 16x16x4 (K=3 padded to 4)
    dim3 gA(NEDGE / 16, NEDGE / 16, BATCH);
    tp_buildA_kernel<<<gA, 32, 0, stream>>>(diff, ei, A);

    // 3) blocked WMMA Cholesky, one workgroup (WGP) per batch
    tp_cholesky_kernel<<<BATCH, 256, 0, stream>>>(A);

    // 4) triangular solves (lam overwrites rhs)
    tp_solve_kernel<<<BATCH, 256, 0, stream>>>(A, rhs);

    // 5) output = v - scatter(2*lam*diff)
    tp_initout_kernel<<<(BATCH * NODES * 3 + 255) / 256, 256, 0, stream>>>(v, out);
    tp_scatter_kernel<<<(BATCH * NEDGE + 255) / 256, 256, 0, stream>>>(rhs, diff, ei, out);
}